// QuConVXYZ_2817498546565
// MI455X (gfx1250) — compile-verified
//
#include <hip/hip_runtime.h>
#include <hip/hip_bf16.h>

// ---------------------------------------------------------------------------
// 20-qubit statevector circuit + partial trace (complex SYRK-like GEMM).
// Gates fused: (Rz*Ry*Rx) -> one 2x2 complex matrix per qubit per layer,
// (Rzz*Ryy*Rxx) -> one 4x4 complex matrix per adjacent pair.
// Partial trace: rdm[a,c] = sum_t psi[t,a]*conj(psi[t,c]) done with
// v_wmma_f32_16x16x4_f32 (fp32 in / fp32 acc == reference precision class).
// ---------------------------------------------------------------------------

#define NQ   20
#define DIM  (1 << NQ)          // 2^20 amplitudes per batch
#define BATCH 4
#define RDIM 1024               // 2^10 (dimA = dimB = 10)
#define W_MUL 0.63245553203367586f  // sqrt(2/5)

struct cplx { float re, im; };
__device__ __forceinline__ cplx cmul(cplx a, cplx b) {
    return { a.re * b.re - a.im * b.im, a.re * b.im + a.im * b.re };
}
__device__ __forceinline__ cplx cadd(cplx a, cplx b) {
    return { a.re + b.re, a.im + b.im };
}

typedef float v2f __attribute__((ext_vector_type(2)));
typedef float v8f __attribute__((ext_vector_type(8)));

// ---------------------------------------------------------------------------
// Gate-matrix preparation: 40 fused 1q gates (2x2 cplx) + 19 fused 2q (4x4).
// gbuf layout: [0, 320) floats = 40 gates * 4 cplx; [320, 928) = 19 * 16 cplx.
// ---------------------------------------------------------------------------
__global__ void prep_gates(const float* __restrict__ weight,
                           float* __restrict__ gbuf) {
    int j = threadIdx.x;
    if (j < 40) {
        int q = j % 20;
        int base = (j < 20) ? 3 * q : (117 + 3 * q);   // layer1 / layer2
        float ax = 0.5f * weight[base]     * W_MUL;
        float ay = 0.5f * weight[base + 1] * W_MUL;
        float az = 0.5f * weight[base + 2] * W_MUL;
        float cx = __cosf(ax), sx = __sinf(ax);
        float cy = __cosf(ay), sy = __sinf(ay);
        float cz = __cosf(az), sz = __sinf(az);
        // M = Ry * Rx
        cplx m00 = { cy * cx,  sy * sx };
        cplx m01 = { -sy * cx, -cy * sx };
        cplx m10 = { sy * cx,  -cy * sx };
        cplx m11 = { cy * cx,  -sy * sx };
        // G = Rz * M  (row0 *= e^{-i az}, row1 *= e^{+i az})
        cplx e0 = { cz, -sz }, e1 = { cz, sz };
        cplx g[4] = { cmul(e0, m00), cmul(e0, m01),
                      cmul(e1, m10), cmul(e1, m11) };
        float* o = gbuf + j * 8;
        for (int k = 0; k < 4; k++) { o[2 * k] = g[k].re; o[2 * k + 1] = g[k].im; }
    } else if (j < 59) {
        int q = j - 40;
        int base = 60 + 3 * q;
        cplx Mx[4][4], My[4][4], Mz[4][4], T[4][4], G[4][4];
        for (int r = 0; r < 4; r++)
            for (int c = 0; c < 4; c++) {
                Mx[r][c] = { 0.f, 0.f }; My[r][c] = { 0.f, 0.f }; Mz[r][c] = { 0.f, 0.f };
            }
        // Rxx = c*I - i s * XX
        float a = 0.5f * weight[base] * W_MUL;
        float c = __cosf(a), s = __sinf(a);
        for (int r = 0; r < 4; r++) Mx[r][r] = { c, 0.f };
        Mx[0][3] = { 0.f, -s }; Mx[1][2] = { 0.f, -s };
        Mx[2][1] = { 0.f, -s }; Mx[3][0] = { 0.f, -s };
        // Ryy = c*I - i s * YY ; YY = [[0,0,0,-1],[0,0,1,0],[0,1,0,0],[-1,0,0,0]]
        a = 0.5f * weight[base + 1] * W_MUL;
        c = __cosf(a); s = __sinf(a);
        for (int r = 0; r < 4; r++) My[r][r] = { c, 0.f };
        My[0][3] = { 0.f,  s }; My[1][2] = { 0.f, -s };
        My[2][1] = { 0.f, -s }; My[3][0] = { 0.f,  s };
        // Rzz = diag(e^{-ia}, e^{ia}, e^{ia}, e^{-ia})
        a = 0.5f * weight[base + 2] * W_MUL;
        c = __cosf(a); s = __sinf(a);
        Mz[0][0] = { c, -s }; Mz[1][1] = { c, s };
        Mz[2][2] = { c, s };  Mz[3][3] = { c, -s };
        // G = Mz * (My * Mx)
        for (int r = 0; r < 4; r++)
            for (int cc = 0; cc < 4; cc++) {
                cplx acc = { 0.f, 0.f };
                for (int k = 0; k < 4; k++) acc = cadd(acc, cmul(My[r][k], Mx[k][cc]));
                T[r][cc] = acc;
            }
        for (int r = 0; r < 4; r++)
            for (int cc = 0; cc < 4; cc++) {
                cplx acc = { 0.f, 0.f };
                for (int k = 0; k < 4; k++) acc = cadd(acc, cmul(Mz[r][k], T[k][cc]));
                G[r][cc] = acc;
            }
        float* o = gbuf + 320 + q * 32;
        for (int r = 0; r < 4; r++)
            for (int cc = 0; cc < 4; cc++) {
                o[2 * (4 * r + cc)]     = G[r][cc].re;
                o[2 * (4 * r + cc) + 1] = G[r][cc].im;
            }
    }
}

// ---------------------------------------------------------------------------
// psi init: complex state from real input x  (batch*2^20 threads)
// ---------------------------------------------------------------------------
__global__ void init_psi(const float* __restrict__ x, float2* __restrict__ psi) {
    unsigned tid = blockIdx.x * blockDim.x + threadIdx.x;
    psi[tid] = make_float2(x[tid], 0.0f);
}

// ---------------------------------------------------------------------------
// Fused 1-qubit gate on bit `bit`.  4*2^19 threads (one per amplitude pair).
// ---------------------------------------------------------------------------
__global__ void apply1q(float2* __restrict__ psi,
                        const float* __restrict__ gbuf, int gate, int bit) {
    const float* g = gbuf + gate * 8;
    cplx g00 = { g[0], g[1] }, g01 = { g[2], g[3] };
    cplx g10 = { g[4], g[5] }, g11 = { g[6], g[7] };
    unsigned tid = blockIdx.x * blockDim.x + threadIdx.x;   // 2^21 total
    unsigned b = tid >> (NQ - 1);
    unsigned p = tid & ((1u << (NQ - 1)) - 1);
    unsigned mask = (1u << bit) - 1;
    unsigned i0 = (b << NQ) | ((p & ~mask) << 1) | (p & mask);
    unsigned i1 = i0 | (1u << bit);
    float2 v0 = psi[i0], v1 = psi[i1];
    cplx a0 = { v0.x, v0.y }, a1 = { v1.x, v1.y };
    cplx r0 = cadd(cmul(g00, a0), cmul(g01, a1));
    cplx r1 = cadd(cmul(g10, a0), cmul(g11, a1));
    psi[i0] = make_float2(r0.re, r0.im);
    psi[i1] = make_float2(r1.re, r1.im);
}

// ---------------------------------------------------------------------------
// Fused 2-qubit gate on bits (bl+1, bl).  4*2^18 threads (one per quad).
// Group index g = 2*bit_high + bit_low matches the 4x4 matrix ordering.
// ---------------------------------------------------------------------------
__global__ void apply2q(float2* __restrict__ psi,
                        const float* __restrict__ gbuf, int gate, int bl) {
    const float* g = gbuf + 320 + gate * 32;
    cplx G[4][4];
    for (int r = 0; r < 4; r++)
        for (int c = 0; c < 4; c++)
            G[r][c] = { g[2 * (4 * r + c)], g[2 * (4 * r + c) + 1] };
    unsigned tid = blockIdx.x * blockDim.x + threadIdx.x;   // 2^20 total
    unsigned b = tid >> (NQ - 2);
    unsigned p = tid & ((1u << (NQ - 2)) - 1);
    unsigned mask = (1u << bl) - 1;
    unsigned i00 = (b << NQ) | ((p & ~mask) << 2) | (p & mask);
    unsigned sl = 1u << bl, sh = 2u << bl;
    unsigned idx[4] = { i00, i00 | sl, i00 | sh, i00 | sh | sl };
    cplx v[4], o[4];
    for (int k = 0; k < 4; k++) {
        float2 t = psi[idx[k]];
        v[k] = { t.x, t.y };
    }
    for (int r = 0; r < 4; r++) {
        cplx acc = cmul(G[r][0], v[0]);
        acc = cadd(acc, cmul(G[r][1], v[1]));
        acc = cadd(acc, cmul(G[r][2], v[2]));
        acc = cadd(acc, cmul(G[r][3], v[3]));
        o[r] = acc;
    }
    for (int k = 0; k < 4; k++)
        psi[idx[k]] = make_float2(o[k].re, o[k].im);
}

// ---------------------------------------------------------------------------
// Partial trace as complex GEMM with fp32 WMMA:
//   rdm[a,c] = sum_t psi[t,a]*conj(psi[t,c])
//   re = R^T R + I^T I ; im = I^T R + (-R)^T I
// Block = 128 threads (4 waves); each wave owns a 32x32 output tile.
// ---------------------------------------------------------------------------
__device__ __forceinline__ v8f wmma_f32(v2f a, v2f b, v8f c) {
#if __has_builtin(__builtin_amdgcn_wmma_f32_16x16x4_f32)
    return __builtin_amdgcn_wmma_f32_16x16x4_f32(false, a, false, b,
                                                 (short)0, c, false, false);
#else
    // compile-safe fallback (flags itself via wmma==0 in histogram)
    for (int k = 0; k < 8; k++) c[k] += a[0] * b[0];
    return c;
#endif
}

__global__ __launch_bounds__(128) void rdm_gemm(const float2* __restrict__ psi,
                                                float2* __restrict__ out) {
    int lane = threadIdx.x & 31;
    int wave = threadIdx.x >> 5;
    int b = blockIdx.z;
    int m0 = blockIdx.y * 64 + (wave >> 1) * 32;  // 'a' rows
    int n0 = blockIdx.x * 64 + (wave & 1) * 32;   // 'c' cols
    const float2* P = psi + (size_t)b * (RDIM * (size_t)RDIM);

    int half = lane >> 4;        // 0: K=0/1, 1: K=2/3
    int csub = lane & 15;

    v8f acc_re[2][2], acc_im[2][2];
    for (int i = 0; i < 2; i++)
        for (int j = 0; j < 2; j++)
            for (int k = 0; k < 8; k++) { acc_re[i][j][k] = 0.f; acc_im[i][j][k] = 0.f; }

    for (int t0 = 0; t0 < RDIM; t0 += 4) {
        int r = t0 + (half << 1);
        v2f aR[2], aI[2], aRn[2], bR[2], bI[2];
#pragma unroll
        for (int i = 0; i < 2; i++) {
            float2 p0 = P[(size_t)r * RDIM + (m0 + i * 16 + csub)];
            float2 p1 = P[(size_t)(r + 1) * RDIM + (m0 + i * 16 + csub)];
            aR[i].x = p0.x; aR[i].y = p1.x;
            aI[i].x = p0.y; aI[i].y = p1.y;
            aRn[i].x = -p0.x; aRn[i].y = -p1.x;
        }
#pragma unroll
        for (int j = 0; j < 2; j++) {
            float2 p0 = P[(size_t)r * RDIM + (n0 + j * 16 + csub)];
            float2 p1 = P[(size_t)(r + 1) * RDIM + (n0 + j * 16 + csub)];
            bR[j].x = p0.x; bR[j].y = p1.x;
            bI[j].x = p0.y; bI[j].y = p1.y;
        }
#pragma unroll
        for (int i = 0; i < 2; i++)
#pragma unroll
            for (int j = 0; j < 2; j++) {
                acc_re[i][j] = wmma_f32(aR[i],  bR[j], acc_re[i][j]);
                acc_re[i][j] = wmma_f32(aI[i],  bI[j], acc_re[i][j]);
                acc_im[i][j] = wmma_f32(aI[i],  bR[j], acc_im[i][j]);
                acc_im[i][j] = wmma_f32(aRn[i], bI[j], acc_im[i][j]);
            }
    }

    // C/D layout: VGPR v -> row m0+v (lanes 0-15) or m0+v+8 (lanes 16-31)
    float2* O = out + (size_t)b * (RDIM * (size_t)RDIM);
#pragma unroll
    for (int i = 0; i < 2; i++)
#pragma unroll
        for (int j = 0; j < 2; j++)
#pragma unroll
            for (int v = 0; v < 8; v++) {
                int row = m0 + i * 16 + v + (half << 3);
                int col = n0 + j * 16 + csub;
                O[(size_t)row * RDIM + col] =
                    make_float2(acc_re[i][j][v], acc_im[i][j][v]);
            }
}

// ---------------------------------------------------------------------------
extern "C" void kernel_launch(void* const* d_in, const int* in_sizes, int n_in,
                              void* d_out, int out_size, void* d_ws, size_t ws_size,
                              hipStream_t stream) {
    const float* x      = (const float*)d_in[0];   // (4, 1, 2^20) float32
    const float* weight = (const float*)d_in[1];   // (177,) float32
    // d_in[2] = dimA (always 10 for this problem)

    float2* psi  = (float2*)d_ws;                                  // 32 MB
    float*  gbuf = (float*)((char*)d_ws +
                            sizeof(float2) * (size_t)BATCH * DIM); // ~4 KB

    prep_gates<<<1, 64, 0, stream>>>(weight, gbuf);
    init_psi<<<(BATCH * DIM) / 256, 256, 0, stream>>>(x, psi);

    // layer 1: qubit q lives on bit 19-q
    for (int q = 0; q < NQ; q++)
        apply1q<<<(BATCH * (DIM / 2)) / 256, 256, 0, stream>>>(psi, gbuf, q, NQ - 1 - q);
    // entangling: qubits (q, q+1) -> bits (19-q, 18-q); low bit = 18-q
    for (int q = 0; q < NQ - 1; q++)
        apply2q<<<(BATCH * (DIM / 4)) / 256, 256, 0, stream>>>(psi, gbuf, q, NQ - 2 - q);
    // layer 2
    for (int q = 0; q < NQ; q++)
        apply1q<<<(BATCH * (DIM / 2)) / 256, 256, 0, stream>>>(psi, gbuf, 20 + q, NQ - 1 - q);

    dim3 grid(RDIM / 64, RDIM / 64, BATCH);
    rdm_gemm<<<grid, 128, 0, stream>>>(psi, (float2*)d_out);
}